// ConsistencyLoss_26680336843577
// MI455X (gfx1250) — compile-verified
//
#include <hip/hip_runtime.h>
#include <hip/hip_bf16.h>
#include <cstdint>

// ConsistencyLoss on MI455X (gfx1250).
// Roofline: 2 streaming passes over features (2 x 204.8 MB) -> ~17.6 us at 23.3 TB/s.
// FLOPs ~2.6e8 -> pure data movement problem; WMMA inapplicable (no matmul structure).
// Pass 1: chunked LDS segment-sum (64 groups x 256 dims = 64 KB accumulator per block,
//         ds_add_f32 at LDS bank rate) -> global FP32 atomics cut 51.2M -> ~8.2M coalesced.
// Pass 2: GLOBAL_LOAD_ASYNC_TO_LDS_B128 double-buffered row staging + s_wait_asynccnt,
//         wave32 __shfl_xor reduction, one global atomic per row.

#define WPB    8      // waves per block (256 threads, wave32)
#define DIM    256    // feature dim (in_sizes[0]/in_sizes[1] == 256)
#define CH_G   64     // groups per LDS chunk (64*256*4B = 64 KB accumulator)
#define NSLICE 32     // row slices per chunk -> 512 blocks = 4 blocks/WGP (LDS-limited)

// ---- CDNA5 async copy helpers (ASYNCcnt-tracked, CDNA5 ISA) ----

__device__ __forceinline__ void async_row_to_lds(uint32_t lds_byte, const float* gptr, int lane) {
    // 32 lanes x 16B = 512B per instruction; two instructions move one 1KB row.
    // INST_OFFSET is added to BOTH global and LDS addresses, so offset:512 reuses VGPRs.
    uint64_t ga = (uint64_t)(uintptr_t)gptr + (uint64_t)(lane << 4);
    uint32_t la = lds_byte + (uint32_t)(lane << 4);
    asm volatile(
        "global_load_async_to_lds_b128 %0, %1, off\n"
        "global_load_async_to_lds_b128 %0, %1, off offset:512"
        :: "v"(la), "v"(ga)
        : "memory");
}
__device__ __forceinline__ void wait_async_le2() {
    asm volatile("s_wait_asynccnt 0x2" ::: "memory");
}
__device__ __forceinline__ void wait_async_le0() {
    asm volatile("s_wait_asynccnt 0x0" ::: "memory");
}

// ---- workspace layout (offsets computed on-device from *gp; G is a device scalar) ----
// ws[0 .. G*DIM)          : per-group feature sums
// ws[G*DIM .. G*DIM+G)    : per-group counts (float)
// ws[G*DIM+G .. G*DIM+2G) : per-group sum of row norms

__global__ __launch_bounds__(256) void k_zero_ws(float* __restrict__ ws,
                                                 const int* __restrict__ gp) {
    int G = *gp;
    size_t total = (size_t)G * DIM + 2u * (size_t)G;
    for (size_t i = (size_t)blockIdx.x * blockDim.x + threadIdx.x; i < total;
         i += (size_t)gridDim.x * blockDim.x)
        ws[i] = 0.0f;
}

// Pass 1: segment sums + counts via chunked LDS accumulation.
// blockIdx.x = row slice, blockIdx.y = group chunk (looped for arbitrary G).
__global__ __launch_bounds__(256) void k_accum(const float* __restrict__ feat,
                                               const int* __restrict__ ids,
                                               const int* __restrict__ gp,
                                               float* __restrict__ ws, int N) {
    int G = *gp;
    float* sums   = ws;
    float* counts = ws + (size_t)G * DIM;

    __shared__ float acc[CH_G * DIM];    // 64 KB group-chunk accumulator
    __shared__ float accCnt[CH_G];

    const int lane = threadIdx.x & 31;
    const int wave = threadIdx.x >> 5;
    const long nbatch = ((long)N + 31) >> 5;   // 32-row id batches

    for (int cbase = blockIdx.y * CH_G; cbase < G; cbase += CH_G * gridDim.y) {
        int chg = min(CH_G, G - cbase);

        for (int i = threadIdx.x; i < CH_G * DIM; i += 256) acc[i] = 0.0f;
        for (int i = threadIdx.x; i < CH_G; i += 256)       accCnt[i] = 0.0f;
        __syncthreads();

        // Each wave scans batches of 32 consecutive rows; lanes test 32 ids at once,
        // wave32 ballot extracts the rows belonging to this chunk.
        for (long b = (long)blockIdx.x * WPB + wave; b < nbatch;
             b += (long)gridDim.x * WPB) {
            long r0 = b << 5;
            long r  = r0 + lane;
            int  g   = (r < N) ? ids[r] : -1;
            int  rel = g - cbase;
            bool m   = (rel >= 0) && (rel < chg);
            if (m) __builtin_prefetch(feat + r * (long)DIM, 0, 0);  // global_prefetch_b8
            unsigned long long mask = __ballot(m);
            while (mask) {
                int j = __ffsll(mask) - 1;
                mask &= mask - 1;
                int grel = __shfl(rel, j, 32);
                const float* row = feat + (r0 + j) * (long)DIM;
                float* arow = acc + grel * DIM;
                #pragma unroll
                for (int k = 0; k < 8; ++k) {
                    int c = lane + 32 * k;        // coalesced load, conflict-free ds_add
                    unsafeAtomicAdd(&arow[c], row[c]);
                }
                if (lane == 0) unsafeAtomicAdd(&accCnt[grel], 1.0f);
            }
        }
        __syncthreads();

        // Flush chunk partials: coalesced global_atomic_add_f32 (NSLICE-way reduction).
        for (int i = threadIdx.x; i < chg * DIM; i += 256)
            unsafeAtomicAdd(&sums[(size_t)cbase * DIM + i], acc[i]);
        for (int i = threadIdx.x; i < chg; i += 256)
            unsafeAtomicAdd(&counts[cbase + i], accCnt[i]);
        __syncthreads();
    }
}

// Pass 2: per-row ||x - center||, center computed on the fly from L2-hot sums/counts.
// Feature rows streamed through LDS with double-buffered async DMA.
__global__ __launch_bounds__(256) void k_norm(const float* __restrict__ feat,
                                              const int* __restrict__ ids,
                                              const int* __restrict__ gp,
                                              float* __restrict__ ws, int N) {
    int G = *gp;
    const float* sums   = ws;
    const float* counts = ws + (size_t)G * DIM;
    float* norm_sum     = ws + (size_t)G * DIM + (size_t)G;

    __shared__ __align__(16) float stage[WPB][2][DIM];

    int lane = threadIdx.x & 31;
    int wave = threadIdx.x >> 5;
    long wid  = (long)blockIdx.x * WPB + wave;
    long wstr = (long)gridDim.x * WPB;

    uint32_t lb[2] = { (uint32_t)(uintptr_t)&stage[wave][0][0],
                       (uint32_t)(uintptr_t)&stage[wave][1][0] };

    long r = wid;
    if (r < N) async_row_to_lds(lb[0], feat + r * (long)DIM, lane);
    int cur = 0;
    for (; r < N; r += wstr) {                 // r wave-uniform -> uniform branches
        long rn = r + wstr;
        if (rn < N) { async_row_to_lds(lb[cur ^ 1], feat + rn * (long)DIM, lane); wait_async_le2(); }
        else        { wait_async_le0(); }

        int g = ids[r];
        float inv = 1.0f / fmaxf(counts[g], 1.0f);
        const float* srow = sums + (size_t)g * DIM;
        const float* b = &stage[wave][cur][0];
        float acc = 0.0f;
        #pragma unroll
        for (int k = 0; k < 8; ++k) {
            int c = lane + 32 * k;
            float d = b[c] - srow[c] * inv;
            acc = fmaf(d, d, acc);
        }
        #pragma unroll
        for (int off = 16; off > 0; off >>= 1)   // wave32 reduction
            acc += __shfl_xor(acc, off, 32);
        if (lane == 0) {
            float rnorm = (acc > 0.0f) ? sqrtf(acc) : 0.0f;  // safe norm
            unsafeAtomicAdd(&norm_sum[g], rnorm);
        }
        cur ^= 1;
    }
}

// Pass 3: reduce over G groups -> scalar.
__global__ __launch_bounds__(256) void k_final(const int* __restrict__ gp,
                                               const float* __restrict__ ws,
                                               float* __restrict__ out) {
    int G = *gp;
    const float* counts   = ws + (size_t)G * DIM;
    const float* norm_sum = counts + G;

    float cs = 0.0f, us = 0.0f;
    for (int g = threadIdx.x; g < G; g += 256) {
        float c  = counts[g];
        float ns = norm_sum[g];
        cs += (c > 1.0f) ? (ns / c) : 0.0f;   // contrib only when count > 1
        us += (c > 0.0f) ? 1.0f : 0.0f;       // unique-group indicator
    }
    __shared__ float sc[256], su[256];
    sc[threadIdx.x] = cs;
    su[threadIdx.x] = us;
    __syncthreads();
    for (int off = 128; off > 0; off >>= 1) {
        if (threadIdx.x < (unsigned)off) {
            sc[threadIdx.x] += sc[threadIdx.x + off];
            su[threadIdx.x] += su[threadIdx.x + off];
        }
        __syncthreads();
    }
    if (threadIdx.x == 0)
        out[0] = sc[0] / fmaxf(su[0], 1.0f);
}

extern "C" void kernel_launch(void* const* d_in, const int* in_sizes, int n_in,
                              void* d_out, int out_size, void* d_ws, size_t ws_size,
                              hipStream_t stream) {
    const float* feat = (const float*)d_in[0];   // [N, 256] f32
    const int*   ids  = (const int*)d_in[1];     // [N] i32
    const int*   gp   = (const int*)d_in[2];     // scalar G on device
    int N = in_sizes[1];

    float* ws  = (float*)d_ws;
    float* out = (float*)d_out;

    k_zero_ws<<<512, 256, 0, stream>>>(ws, gp);
    k_accum  <<<dim3(NSLICE, 16), 256, 0, stream>>>(feat, ids, gp, ws, N);
    k_norm   <<<1024, 256, 0, stream>>>(feat, ids, gp, ws, N);
    k_final  <<<1,    256, 0, stream>>>(gp, ws, out);
}